// RegionProposalNetwork_38079180046389
// MI455X (gfx1250) — compile-verified
//
#include <hip/hip_runtime.h>
#include <math.h>

typedef float v2f __attribute__((ext_vector_type(2)));
typedef float v8f __attribute__((ext_vector_type(8)));

#define IMG_W 1344.0f
#define IMG_H 800.0f

// ---------- helpers ----------
__device__ __forceinline__ unsigned ordf(float f) {
    unsigned u = __float_as_uint(f);
    return (u & 0x80000000u) ? ~u : (u | 0x80000000u);
}
__device__ __forceinline__ float unordf(unsigned u) {
    unsigned b = (u & 0x80000000u) ? (u & 0x7fffffffu) : ~u;
    return __uint_as_float(b);
}
__device__ __forceinline__ float sigmoidf(float x) {
    if (x >= 0.f) return 1.f / (1.f + expf(-x));
    float e = expf(x);
    return e / (1.f + e);
}

// ---------- weight prep: OIHW -> fragment-ready K-pair-interleaved layout ----------
// dst[kc*8192 + p*512 + n*2 + e]  where k = kc*32 + p*2 + e  (kc chunk, p pair 0..15)
// and source k-order is k = tap*256 + ci  (tap = ky*3+kx).
__global__ void wt_prep(const float* __restrict__ w, float* __restrict__ wt) {
    int i = blockIdx.x * 256 + threadIdx.x;           // 0 .. 2304*256-1
    if (i >= 2304 * 256) return;
    int kc   = i >> 13;                               // /8192
    int rem  = i & 8191;
    int p    = rem >> 9;                              // /512
    int rem2 = rem & 511;
    int n    = rem2 >> 1;
    int e    = rem2 & 1;
    int k    = kc * 32 + p * 2 + e;                   // 0..2303
    int ci   = k & 255;
    int tap  = k >> 8;                                // 0..8  (= ky*3+kx)
    wt[i] = w[n * 2304 + ci * 9 + tap];
}

// ---------- anchors (clipped) ----------
__global__ void anchors_kernel(float* __restrict__ out) {
    int i = blockIdx.x * 256 + threadIdx.x;
    if (i >= 252000) return;
    int H, W, idx;
    float size;
    if (i < 201600) { H = 200; W = 336; size = 32.f; idx = i; }
    else            { H = 100; W = 168; size = 64.f; idx = i - 201600; }
    int HW = H * W;
    int a = idx / HW, rem = idx % HW, y = rem / W, x = rem % W;
    float sw = IMG_W / (float)W, sh = IMG_H / (float)H;
    float ratio = (a == 0) ? 0.5f : ((a == 1) ? 1.f : 2.f);
    float sq = sqrtf(ratio);
    float ah = size * sq, aw = size / sq;
    float cx = (x + 0.5f) * sw, cy = (y + 0.5f) * sh;
    float x1 = cx - aw * 0.5f, y1 = cy - ah * 0.5f;
    float x2 = cx + aw * 0.5f, y2 = cy + ah * 0.5f;
    x1 = fminf(fmaxf(x1, 0.f), IMG_W); x2 = fminf(fmaxf(x2, 0.f), IMG_W);
    y1 = fminf(fmaxf(y1, 0.f), IMG_H); y2 = fminf(fmaxf(y2, 0.f), IMG_H);
    out[(size_t)i * 4 + 0] = x1; out[(size_t)i * 4 + 1] = y1;
    out[(size_t)i * 4 + 2] = x2; out[(size_t)i * 4 + 3] = y2;
}

// ---------- fused 3x3 conv (WMMA f32, async-to-LDS double-buffered) + 1x1 heads ----------
// Grid: HW/32 blocks of 256 threads (8 waves). WG tile: M=32 pixels, N=256 channels.
// K = 2304 in chunks of 32.  Wave (wm, wn) owns M rows [wm*16,+16), N cols [wn*64,+64)
// as 4 accumulators of v_wmma_f32_16x16x4_f32.
// B tiles stream via GLOBAL_LOAD_ASYNC_TO_LDS_B128 (ASYNCcnt) in a fragment-ready,
// bank-conflict-free layout: pair row stride 544 floats (544 mod 64 == 32), so the
// two half-waves of a b64 fragment load land on disjoint bank halves.
__global__ __launch_bounds__(256)
void conv_head_kernel(const float* __restrict__ X,   // [256][HW]
                      const float* __restrict__ WT,  // fragment-ready prepped weights
                      const float* __restrict__ CB,  // [256]
                      const float* __restrict__ DW,  // [3][256]
                      const float* __restrict__ DB,  // [3]
                      const float* __restrict__ RW,  // [12][256]
                      const float* __restrict__ RB,  // [12]
                      float* __restrict__ obj_out,       // level-local: obj_out[idx]
                      float* __restrict__ deltas_out,    // global deltas base
                      unsigned long long* __restrict__ keys, // level-local keys
                      int H, int W, int lvl_off) {
    __shared__ __align__(16) float smem[19584];   // 78.3 KB
    float* sA0 = smem;              // [32][34]   phase 1, buffer 0
    float* sA1 = smem + 1088;       // [32][34]   phase 1, buffer 1
    float* sB0 = smem + 2176;       // [16 pairs][544] phase 1, buffer 0
    float* sB1 = smem + 10880;      // [16 pairs][544] phase 1, buffer 1
    float* sH  = smem;              // [32][257]  phase 2 (overlays phase-1)
    float* sWt = smem + 8224;       // [15][256]  phase 2

    const int t = threadIdx.x;
    const int HW = H * W;
    const int m0 = blockIdx.x * 32;
    const int lane = t & 31;
    const int wv = t >> 5;
    const int wm = wv & 1;          // 0..1 (M)
    const int wn = wv >> 1;         // 0..3 (N)
    const int lh = lane >> 4;       // half-wave select
    const int l15 = lane & 15;
    const int kb = lh * 2;          // K sub-base per ISA A/B layout

    auto asyncB = [&](float* buf, int kc) {
        const float* wp = WT + (size_t)kc * 8192;
        #pragma unroll
        for (int r = 0; r < 8; ++r) {
            int e4 = r * 256 + t;           // b128 index: 128 per pair-row
            int p = e4 >> 7;                // pair row 0..15
            int col = (e4 & 127) * 4;       // float col 0..508, 16B aligned
            unsigned lds = (unsigned)(unsigned long long)(buf + p * 544 + col);
            const float* g = wp + p * 512 + col;
            asm volatile("global_load_async_to_lds_b128 %0, %1, off"
                         :: "v"(lds), "v"(g) : "memory");
        }
    };
    auto loadA = [&](float* buf, int kc) {
        const int k0 = kc * 32;
        const int tap = k0 >> 8;
        const int ci0 = k0 & 255;
        const int ky = tap / 3 - 1;
        const int kx = tap % 3 - 1;
        #pragma unroll
        for (int r = 0; r < 4; ++r) {
            int idx = r * 256 + t;
            int kk = idx >> 5, ml2 = idx & 31;
            int mg = m0 + ml2;
            int y = mg / W, x = mg % W;
            int yy = y + ky, xx = x + kx;
            float v = 0.f;
            if (yy >= 0 && yy < H && xx >= 0 && xx < W)
                v = X[(size_t)(ci0 + kk) * HW + yy * W + xx];
            buf[ml2 * 34 + kk] = v;
        }
    };

    v8f acc[4];
    #pragma unroll
    for (int i = 0; i < 4; ++i) acc[i] = (v8f){0.f,0.f,0.f,0.f,0.f,0.f,0.f,0.f};

    // prologue: stage chunk 0
    asyncB(sB0, 0);
    loadA(sA0, 0);
    asm volatile("s_wait_asynccnt 0x0" ::: "memory");
    __syncthreads();

    const int arow = (wm * 16 + l15) * 34;
    const int nfb  = wn * 64 + l15;

    for (int kc = 0; kc < 72; ++kc) {
        float* curA = (kc & 1) ? sA1 : sA0;
        float* curB = (kc & 1) ? sB1 : sB0;
        float* nxtA = (kc & 1) ? sA0 : sA1;
        float* nxtB = (kc & 1) ? sB0 : sB1;
        const bool has_next = (kc < 71);
        if (has_next) asyncB(nxtB, kc + 1);     // DMA overlaps the WMMAs below

        #pragma unroll
        for (int s = 0; s < 8; ++s) {
            const int kk = s * 4 + kb;
            // A 16x4 fragment: lane=M, contiguous (kk,kk+1) -> one ds_load_b64
            v2f a = *(const v2f*)(curA + arow + kk);
            const float* bpair = curB + (2 * s + lh) * 544;  // fragment-ready row
            #pragma unroll
            for (int nt = 0; nt < 4; ++nt) {
                // B 4x16 fragment: lane=N, interleaved pair -> one ds_load_b64
                v2f b = *(const v2f*)(bpair + (nfb + nt * 16) * 2);
                acc[nt] = __builtin_amdgcn_wmma_f32_16x16x4_f32(
                    false, a, false, b, (short)0, acc[nt], false, false);
            }
        }

        if (has_next) {
            loadA(nxtA, kc + 1);
            asm volatile("s_wait_asynccnt 0x0" ::: "memory");
        }
        __syncthreads();
    }

    // ---- epilogue: h tile (conv + bias) to LDS, head weights to LDS ----
    #pragma unroll
    for (int nt = 0; nt < 4; ++nt) {
        const int nf = wn * 64 + nt * 16 + l15;
        const float bias = CB[nf];
        #pragma unroll
        for (int r = 0; r < 8; ++r) {               // C/D: vgpr r + 8*half = M
            int ml = wm * 16 + r + (lh << 3);
            sH[ml * 257 + nf] = acc[nt][r] + bias;
        }
    }
    #pragma unroll
    for (int o = 0; o < 15; ++o)
        sWt[o * 256 + t] = (o < 3) ? DW[o * 256 + t] : RW[(o - 3) * 256 + t];
    __syncthreads();

    // ---- 1x1 heads: 8 lanes per pixel, 32 channels each, shuffle-reduce ----
    const int ml = t >> 3;      // pixel 0..31
    const int g = t & 7;        // channel group
    float s[15];
    #pragma unroll
    for (int o = 0; o < 15; ++o) s[o] = 0.f;
    for (int c = 0; c < 32; ++c) {
        int ci = g * 32 + c;
        float hv = sH[ml * 257 + ci];
        #pragma unroll
        for (int o = 0; o < 15; ++o) s[o] += sWt[o * 256 + ci] * hv;
    }
    #pragma unroll
    for (int o = 0; o < 15; ++o) {
        s[o] += __shfl_xor(s[o], 1, 32);
        s[o] += __shfl_xor(s[o], 2, 32);
        s[o] += __shfl_xor(s[o], 4, 32);
    }
    if (g == 0) {
        const int mg = m0 + ml;
        #pragma unroll
        for (int a = 0; a < 3; ++a) {
            float sc = sigmoidf(s[a] + DB[a]);
            int idx = a * HW + mg;
            obj_out[idx] = sc;
            float masked = (sc > 0.f) ? sc : -1.f;
            keys[idx] = ((unsigned long long)ordf(masked) << 32) |
                        (unsigned long long)(0xFFFFFFFFu - (unsigned)idx);
        }
        #pragma unroll
        for (int j = 0; j < 12; ++j) {              // j = c*3 + a
            float d = s[3 + j] + RB[j];
            int a = j % 3, c = j / 3;
            int idx = a * HW + mg;
            deltas_out[(size_t)(lvl_off + idx) * 4 + c] = d;
        }
    }
}

// ---------- per-chunk bitonic sort (desc), keep sorted top 2048 ----------
__global__ __launch_bounds__(1024)
void chunk_sort(const unsigned long long* __restrict__ keys,
                unsigned long long* __restrict__ out, int N) {
    __shared__ unsigned long long s[4096];
    const int base = blockIdx.x * 4096;
    for (int r = 0; r < 4; ++r) {
        int i = r * 1024 + threadIdx.x;
        int e = base + i;
        s[i] = (e < N) ? keys[e] : 0ULL;
    }
    __syncthreads();
    for (int k = 2; k <= 4096; k <<= 1) {
        for (int j = k >> 1; j > 0; j >>= 1) {
            for (int r = 0; r < 4; ++r) {
                int i = r * 1024 + threadIdx.x;
                int l = i ^ j;
                if (l > i) {
                    unsigned long long a = s[i], b = s[l];
                    bool sw = ((i & k) == 0) ? (a < b) : (a > b);   // descending
                    if (sw) { s[i] = b; s[l] = a; }
                }
            }
            __syncthreads();
        }
    }
    for (int r = 0; r < 2; ++r) {
        int i = r * 1024 + threadIdx.x;
        out[(size_t)blockIdx.x * 2048 + i] = s[i];
    }
}

// ---------- merge two sorted-desc 2048 runs -> sorted-desc top 2048 ----------
__global__ __launch_bounds__(1024)
void merge_topk(const unsigned long long* __restrict__ src,
                unsigned long long* __restrict__ dst) {
    __shared__ unsigned long long s[4096];
    const unsigned long long* l0 = src + (size_t)(2 * blockIdx.x) * 2048;
    const unsigned long long* l1 = src + (size_t)(2 * blockIdx.x + 1) * 2048;
    for (int r = 0; r < 2; ++r) {
        int i = r * 1024 + threadIdx.x;
        s[i] = l0[i];
        s[4095 - i] = l1[i];            // reversed -> bitonic
    }
    __syncthreads();
    for (int j = 2048; j > 0; j >>= 1) {
        for (int r = 0; r < 4; ++r) {
            int i = r * 1024 + threadIdx.x;
            int l = i ^ j;
            if (l > i) {
                unsigned long long a = s[i], b = s[l];
                if (a < b) { s[i] = b; s[l] = a; }   // descending merge
            }
        }
        __syncthreads();
    }
    for (int r = 0; r < 2; ++r) {
        int i = r * 1024 + threadIdx.x;
        dst[(size_t)blockIdx.x * 2048 + i] = s[i];
    }
}

// ---------- gather + decode + clip + sequential NMS + stable emit ----------
__global__ __launch_bounds__(1024)
void nms_emit(const unsigned long long* __restrict__ cand,
              const float* __restrict__ deltas_base,
              float* __restrict__ fscores, float* __restrict__ fprops,
              int H, int W, float size, int lvl_off) {
    __shared__ float bx1[2000], by1[2000], bx2[2000], by2[2000], bsc[2000];
    __shared__ int keep[2000], rnk[2000];
    __shared__ int nk_sh;
    const int t = threadIdx.x;
    const int HW = H * W;
    const float sw = IMG_W / (float)W, sh = IMG_H / (float)H;

    for (int i = t; i < 2000; i += 1024) {
        unsigned long long key = cand[i];
        float msc = unordf((unsigned)(key >> 32));
        unsigned idx = 0xFFFFFFFFu - (unsigned)(key & 0xFFFFFFFFull);
        int a = idx / HW, rem = idx % HW, y = rem / W, x = rem % W;
        float ratio = (a == 0) ? 0.5f : ((a == 1) ? 1.f : 2.f);
        float sq = sqrtf(ratio);
        float ah = size * sq, aw = size / sq;
        float cx = (x + 0.5f) * sw, cy = (y + 0.5f) * sh;
        float ax1 = cx - aw * 0.5f, ay1 = cy - ah * 0.5f;
        float ax2 = cx + aw * 0.5f, ay2 = cy + ah * 0.5f;
        ax1 = fminf(fmaxf(ax1, 0.f), IMG_W); ax2 = fminf(fmaxf(ax2, 0.f), IMG_W);
        ay1 = fminf(fmaxf(ay1, 0.f), IMG_H); ay2 = fminf(fmaxf(ay2, 0.f), IMG_H);
        float aww = ax2 - ax1, ahh = ay2 - ay1;
        float acx = ay1 + aww * 0.5f;   // reference's x/y swap, kept faithfully
        float acy = ax1 + ahh * 0.5f;
        const float* dp = deltas_base + (size_t)(lvl_off + idx) * 4;
        float dx = dp[0], dy = dp[1], dwv = dp[2], dhv = dp[3];
        float px = acx + dx * aww, py = acy + dy * ahh;
        float pw = aww * expf(dwv), ph = ahh * expf(dhv);
        float x1 = px - pw * 0.5f, y1 = py - ph * 0.5f;
        float x2 = px + pw * 0.5f, y2 = py + ph * 0.5f;
        x1 = fminf(fmaxf(x1, 0.f), IMG_W); x2 = fminf(fmaxf(x2, 0.f), IMG_W);
        y1 = fminf(fmaxf(y1, 0.f), IMG_H); y2 = fminf(fmaxf(y2, 0.f), IMG_H);
        bx1[i] = x1; by1[i] = y1; bx2[i] = x2; by2[i] = y2;
        bsc[i] = msc;
        keep[i] = (msc > 0.f) ? 1 : 0;
    }
    __syncthreads();

    for (int i = 0; i < 2000; ++i) {           // matches lax.scan order exactly
        if (keep[i]) {
            float ax1 = bx1[i], ay1 = by1[i], ax2 = bx2[i], ay2 = by2[i];
            float areaI = (ax2 - ax1) * (ay2 - ay1);
            for (int j = i + 1 + t; j < 2000; j += 1024) {
                if (keep[j]) {
                    float ix1 = fmaxf(ax1, bx1[j]), iy1 = fmaxf(ay1, by1[j]);
                    float ix2 = fminf(ax2, bx2[j]), iy2 = fminf(ay2, by2[j]);
                    float iw = fmaxf(ix2 - ix1, 0.f), ih = fmaxf(iy2 - iy1, 0.f);
                    float inter = iw * ih;
                    float areaJ = (bx2[j] - bx1[j]) * (by2[j] - by1[j]);
                    float iou = inter / fmaxf(areaI + areaJ - inter, 1e-9f);
                    if (iou > 0.7f) keep[j] = 0;
                }
            }
        }
        __syncthreads();
    }

    if (t == 0) {
        int kk = 0, nn = 0;
        for (int i = 0; i < 2000; ++i) rnk[i] = keep[i] ? kk++ : nn++;
        nk_sh = kk;
    }
    __syncthreads();
    const int nk = nk_sh;
    for (int i = t; i < 2000; i += 1024) {
        int slot = keep[i] ? rnk[i] : nk + rnk[i];
        if (slot < 1000) {
            if (keep[i]) {
                fscores[slot] = bsc[i];
                fprops[slot * 4 + 0] = bx1[i]; fprops[slot * 4 + 1] = by1[i];
                fprops[slot * 4 + 2] = bx2[i]; fprops[slot * 4 + 3] = by2[i];
            } else {
                fscores[slot] = 0.f;
                fprops[slot * 4 + 0] = 0.f; fprops[slot * 4 + 1] = 0.f;
                fprops[slot * 4 + 2] = 0.f; fprops[slot * 4 + 3] = 0.f;
            }
        }
    }
}

// ---------- driver ----------
extern "C" void kernel_launch(void* const* d_in, const int* in_sizes, int n_in,
                              void* d_out, int out_size, void* d_ws, size_t ws_size,
                              hipStream_t stream) {
    (void)in_sizes; (void)n_in; (void)out_size; (void)ws_size;
    const float* fpn0    = (const float*)d_in[0];
    const float* fpn1    = (const float*)d_in[1];
    const float* conv_w0 = (const float*)d_in[2];
    const float* conv_b0 = (const float*)d_in[3];
    const float* conv_w1 = (const float*)d_in[4];
    const float* conv_b1 = (const float*)d_in[5];
    const float* det_w   = (const float*)d_in[6];
    const float* det_b   = (const float*)d_in[7];
    const float* reg_w   = (const float*)d_in[8];
    const float* reg_b   = (const float*)d_in[9];
    float* out = (float*)d_out;

    // workspace layout (~9.4 MB)
    unsigned long long* keys0 = (unsigned long long*)d_ws;          // 201600
    unsigned long long* keys1 = keys0 + 201600;                     // 50400
    float* wt0 = (float*)(keys1 + 50400);                           // 2304*256
    float* wt1 = wt0 + 2304 * 256;
    unsigned long long* candA0 = (unsigned long long*)(wt1 + 2304 * 256);
    unsigned long long* candB0 = candA0 + 64 * 2048;
    unsigned long long* candA1 = candB0 + 64 * 2048;
    unsigned long long* candB1 = candA1 + 16 * 2048;

    // output layout (flat concat, reference return order)
    float* anchors_out = out;                 // 252000*4
    float* obj_out     = out + 1008000;       // 252000
    float* scores_out  = out + 1260000;       // 2000
    float* props_out   = out + 1262000;       // 8000
    float* deltas_out  = out + 1270000;       // 1008000

    wt_prep<<<2304, 256, 0, stream>>>(conv_w0, wt0);
    wt_prep<<<2304, 256, 0, stream>>>(conv_w1, wt1);
    anchors_kernel<<<(252000 + 255) / 256, 256, 0, stream>>>(anchors_out);

    // ---- level 0: H=200 W=336, size=32 ----
    conv_head_kernel<<<2100, 256, 0, stream>>>(fpn0, wt0, conv_b0, det_w, det_b,
                                               reg_w, reg_b, obj_out, deltas_out,
                                               keys0, 200, 336, 0);
    chunk_sort<<<64, 1024, 0, stream>>>(keys0, candA0, 201600);
    {
        unsigned long long* s = candA0; unsigned long long* d = candB0;
        for (int cnt = 64; cnt > 1; cnt >>= 1) {
            merge_topk<<<cnt / 2, 1024, 0, stream>>>(s, d);
            unsigned long long* tmp = s; s = d; d = tmp;
        }
        nms_emit<<<1, 1024, 0, stream>>>(s, deltas_out, scores_out, props_out,
                                         200, 336, 32.f, 0);
    }

    // ---- level 1: H=100 W=168, size=64 ----
    conv_head_kernel<<<525, 256, 0, stream>>>(fpn1, wt1, conv_b1, det_w, det_b,
                                              reg_w, reg_b, obj_out + 201600,
                                              deltas_out, keys1, 100, 168, 201600);
    chunk_sort<<<16, 1024, 0, stream>>>(keys1, candA1, 50400);
    {
        unsigned long long* s = candA1; unsigned long long* d = candB1;
        for (int cnt = 16; cnt > 1; cnt >>= 1) {
            merge_topk<<<cnt / 2, 1024, 0, stream>>>(s, d);
            unsigned long long* tmp = s; s = d; d = tmp;
        }
        nms_emit<<<1, 1024, 0, stream>>>(s, deltas_out, scores_out + 1000,
                                         props_out + 4000, 100, 168, 64.f, 201600);
    }
}